// MoE_77120432767217
// MI455X (gfx1250) — compile-verified
//
#include <hip/hip_runtime.h>
#include <math.h>

typedef float v2f __attribute__((ext_vector_type(2)));
typedef float v8f __attribute__((ext_vector_type(8)));

#define C_    256
#define HID_  512
#define NSP   3136      // 56*56
#define E_    8
#define B_    8
#define G_    4
#define TN    32        // spatial tile (2 WMMA col-tiles)
#define NT    98        // NSP / TN
#define SP    34        // padded LDS row stride (avoids 64-apart bank conflicts)

static __device__ __forceinline__ v8f zero8() { v8f z = {}; return z; }

static __device__ __forceinline__ v8f wmma4(v2f a, v2f b, v8f c) {
  // D = A(16x4 f32) * B(4x16 f32) + C(16x16 f32)
  return __builtin_amdgcn_wmma_f32_16x16x4_f32(false, a, false, b, (short)0, c,
                                               false, false);
}

// ---------------- global average pool: (B,C,HW) -> (B*C,) ----------------
__global__ void gap_kernel(const float* __restrict__ x, float* __restrict__ xgap) {
  __shared__ float red[256];
  const float* p = x + (size_t)blockIdx.x * NSP;
  float s = 0.f;
  for (int i = threadIdx.x; i < NSP; i += 256) s += p[i];
  red[threadIdx.x] = s;
  __syncthreads();
  for (int off = 128; off > 0; off >>= 1) {
    if (threadIdx.x < off) red[threadIdx.x] += red[threadIdx.x + off];
    __syncthreads();
  }
  if (threadIdx.x == 0) xgap[blockIdx.x] = red[0] * (1.0f / (float)NSP);
}

// -------- gating: logits -> softmax -> top2 -> dense cw; plus loss --------
__global__ void gate_kernel(const float* __restrict__ xgap,
                            const float* __restrict__ gates,
                            float* __restrict__ cw,
                            float* __restrict__ loss_out) {
  __shared__ float s_gap[B_ * C_];
  __shared__ float s_val[G_ * B_ * E_];   // logits, then probs
  __shared__ float s_use[E_];
  int tid = threadIdx.x;                  // 256 threads == G*B*E
  for (int i = tid; i < B_ * C_; i += 256) s_gap[i] = xgap[i];
  __syncthreads();
  int gg = tid >> 6, b = (tid >> 3) & 7, e = tid & 7;
  float acc = 0.f;
  for (int c = 0; c < C_; ++c)
    acc += s_gap[b * C_ + c] * gates[(gg * C_ + c) * E_ + e];
  s_val[tid] = acc;
  __syncthreads();
  int base = tid & ~7;
  float mx = -3.4e38f;
  for (int j = 0; j < 8; ++j) mx = fmaxf(mx, s_val[base + j]);
  float se = 0.f;
  for (int j = 0; j < 8; ++j) se += expf(s_val[base + j] - mx);
  float p = expf(acc - mx) / se;
  __syncthreads();
  s_val[tid] = p;                         // now probs
  __syncthreads();
  // top-2 (first-occurrence tie break, matching lax.top_k)
  float p1 = -1.f; int i1 = 0;
  for (int j = 0; j < 8; ++j) { float pj = s_val[base + j]; if (pj > p1) { p1 = pj; i1 = j; } }
  float p2 = -1.f; int i2 = -1;
  for (int j = 0; j < 8; ++j) {
    if (j == i1) continue;
    float pj = s_val[base + j];
    if (pj > p2) { p2 = pj; i2 = j; }
  }
  float s = p1 + p2 + 1e-10f;
  cw[tid] = (e == i1) ? (p1 / s) : ((e == i2) ? (p2 / s) : 0.f);
  // load-balance loss: cv^2 of usage (unbiased var)
  if (tid < E_) {
    float u = 0.f;
    for (int k = 0; k < G_ * B_; ++k) u += s_val[k * E_ + tid];
    s_use[tid] = u * (1.f / (float)(G_ * B_));
  }
  __syncthreads();
  if (tid == 0) {
    float mean = 0.f;
    for (int j = 0; j < E_; ++j) mean += s_use[j];
    mean *= (1.f / (float)E_);
    float var = 0.f;
    for (int j = 0; j < E_; ++j) { float d = s_use[j] - mean; var += d * d; }
    var *= (1.f / (float)(E_ - 1));
    loss_out[0] = var / (mean * mean + 1e-10f);
  }
}

// ---- fold BN into per-(e,hid) affine: h*sc + ((b2-mean)*sc + beta) ----
__global__ void bn_affine_kernel(const float* __restrict__ b2,
                                 const float* __restrict__ g,
                                 const float* __restrict__ be,
                                 const float* __restrict__ m,
                                 const float* __restrict__ v,
                                 float* __restrict__ A2, float* __restrict__ B2c) {
  int i = blockIdx.x * 256 + threadIdx.x;   // < E*HID = 4096
  float sc = g[i] * rsqrtf(v[i] + 1e-5f);
  A2[i] = sc;
  B2c[i] = (b2[i] - m[i]) * sc + be[i];
}

// -------------------- fused per-tile MoE expert pipeline --------------------
// grid: B*NT blocks of 256 threads (8 wave32s). Per block: one (b, 32-col) tile,
// loop all 8 experts through 3 WMMA GEMM stages in LDS, accumulate 4 gate
// outputs in VGPRs, single store at the end.
__global__ __launch_bounds__(256) void moe_main(
    const float* __restrict__ x,  const float* __restrict__ W1,
    const float* __restrict__ b1, const float* __restrict__ W2,
    const float* __restrict__ W3, const float* __restrict__ b3,
    const float* __restrict__ cw, const float* __restrict__ A2,
    const float* __restrict__ B2c, float* __restrict__ out) {
  __shared__ float bufA[HID_ * SP];   // h1
  __shared__ float bufB[HID_ * SP];   // x tile (rows 0..255) then h2
  const int b   = blockIdx.x / NT;
  const int n0  = (blockIdx.x % NT) * TN;
  const int tid = threadIdx.x;
  const int wave = tid >> 5;
  const int lane = tid & 31;
  const int lm = lane & 15;         // M/N index within tile
  const int lh = lane >> 4;         // K/M half select

  v8f oacc[G_][2][2];
#pragma unroll
  for (int g = 0; g < G_; ++g)
#pragma unroll
    for (int i = 0; i < 2; ++i)
#pragma unroll
      for (int ct = 0; ct < 2; ++ct) oacc[g][i][ct] = zero8();

  const float* xb = x + (size_t)b * C_ * NSP;

  for (int e = 0; e < E_; ++e) {
    float cwg[G_];
#pragma unroll
    for (int g = 0; g < G_; ++g) cwg[g] = cw[(g * B_ + b) * E_ + e];

    __syncthreads();   // previous expert's stage-3 reads of bufB done
    // stage x tile (256 x 32) into bufB — L2-resident after first pass
    for (int idx = tid; idx < C_ * TN; idx += 256) {
      int r = idx >> 5, c = idx & 31;
      bufB[r * SP + c] = xb[(size_t)r * NSP + n0 + c];
    }
    __syncthreads();

    // ---------------- stage 1: h1 = W1[e] @ x + b1 ----------------
    {
      const float* W1e = W1 + (size_t)e * HID_ * C_;
      v8f acc[4][2];
#pragma unroll
      for (int i = 0; i < 4; ++i) { acc[i][0] = zero8(); acc[i][1] = zero8(); }
      for (int k0 = 0; k0 < C_; k0 += 4) {
        v2f bf[2];
#pragma unroll
        for (int ct = 0; ct < 2; ++ct) {
          const float* lp = &bufB[(k0 + 2 * lh) * SP + ct * 16 + lm];
          bf[ct].x = lp[0]; bf[ct].y = lp[SP];
        }
#pragma unroll
        for (int i = 0; i < 4; ++i) {
          int m0 = wave * 64 + i * 16;
          const float* ap = W1e + (size_t)(m0 + lm) * C_ + k0 + 2 * lh;
          v2f a; a.x = ap[0]; a.y = ap[1];
          acc[i][0] = wmma4(a, bf[0], acc[i][0]);
          acc[i][1] = wmma4(a, bf[1], acc[i][1]);
        }
      }
      const float* b1e = b1 + e * HID_;
#pragma unroll
      for (int i = 0; i < 4; ++i)
#pragma unroll
        for (int ct = 0; ct < 2; ++ct)
#pragma unroll
          for (int v = 0; v < 8; ++v) {
            int row = wave * 64 + i * 16 + v + 8 * lh;
            bufA[row * SP + ct * 16 + lm] = acc[i][ct][v] + b1e[row];
          }
    }
    __syncthreads();

    // -------- stage 2: h2 = relu((W2[e] @ h1) * sc + bias') --------
    {
      const float* W2e = W2 + (size_t)e * HID_ * HID_;
      v8f acc[4][2];
#pragma unroll
      for (int i = 0; i < 4; ++i) { acc[i][0] = zero8(); acc[i][1] = zero8(); }
      for (int k0 = 0; k0 < HID_; k0 += 4) {
        v2f bf[2];
#pragma unroll
        for (int ct = 0; ct < 2; ++ct) {
          const float* lp = &bufA[(k0 + 2 * lh) * SP + ct * 16 + lm];
          bf[ct].x = lp[0]; bf[ct].y = lp[SP];
        }
#pragma unroll
        for (int i = 0; i < 4; ++i) {
          int m0 = wave * 64 + i * 16;
          const float* ap = W2e + (size_t)(m0 + lm) * HID_ + k0 + 2 * lh;
          v2f a; a.x = ap[0]; a.y = ap[1];
          acc[i][0] = wmma4(a, bf[0], acc[i][0]);
          acc[i][1] = wmma4(a, bf[1], acc[i][1]);
        }
      }
      const float* A2e = A2 + e * HID_;
      const float* B2e = B2c + e * HID_;
#pragma unroll
      for (int i = 0; i < 4; ++i)
#pragma unroll
        for (int ct = 0; ct < 2; ++ct)
#pragma unroll
          for (int v = 0; v < 8; ++v) {
            int row = wave * 64 + i * 16 + v + 8 * lh;
            float hv = fmaxf(acc[i][ct][v] * A2e[row] + B2e[row], 0.f);
            bufB[row * SP + ct * 16 + lm] = hv;
          }
    }
    __syncthreads();

    // ------ stage 3: y = W3[e] @ h2 + b3 ; oacc[g] += cw[g,b,e]*y ------
    {
      const float* W3e = W3 + (size_t)e * C_ * HID_;
      v8f acc[2][2];
#pragma unroll
      for (int i = 0; i < 2; ++i) { acc[i][0] = zero8(); acc[i][1] = zero8(); }
      for (int k0 = 0; k0 < HID_; k0 += 4) {
        v2f bf[2];
#pragma unroll
        for (int ct = 0; ct < 2; ++ct) {
          const float* lp = &bufB[(k0 + 2 * lh) * SP + ct * 16 + lm];
          bf[ct].x = lp[0]; bf[ct].y = lp[SP];
        }
#pragma unroll
        for (int i = 0; i < 2; ++i) {
          int m0 = wave * 32 + i * 16;
          const float* ap = W3e + (size_t)(m0 + lm) * HID_ + k0 + 2 * lh;
          v2f a; a.x = ap[0]; a.y = ap[1];
          acc[i][0] = wmma4(a, bf[0], acc[i][0]);
          acc[i][1] = wmma4(a, bf[1], acc[i][1]);
        }
      }
      const float* b3e = b3 + e * C_;
#pragma unroll
      for (int i = 0; i < 2; ++i)
#pragma unroll
        for (int ct = 0; ct < 2; ++ct)
#pragma unroll
          for (int v = 0; v < 8; ++v) {
            int row = wave * 32 + i * 16 + v + 8 * lh;
            float y = acc[i][ct][v] + b3e[row];
#pragma unroll
            for (int g = 0; g < G_; ++g) oacc[g][i][ct][v] += cwg[g] * y;
          }
    }
  }

  // single store of all 4 gated outputs
#pragma unroll
  for (int g = 0; g < G_; ++g)
#pragma unroll
    for (int i = 0; i < 2; ++i)
#pragma unroll
      for (int ct = 0; ct < 2; ++ct)
#pragma unroll
        for (int v = 0; v < 8; ++v) {
          int row = wave * 32 + i * 16 + v + 8 * lh;
          out[(((size_t)g * B_ + b) * C_ + row) * NSP + n0 + ct * 16 + lm] =
              oacc[g][i][ct][v];
        }
}

extern "C" void kernel_launch(void* const* d_in, const int* in_sizes, int n_in,
                              void* d_out, int out_size, void* d_ws, size_t ws_size,
                              hipStream_t stream) {
  const float* x    = (const float*)d_in[0];
  const float* gat  = (const float*)d_in[1];
  const float* W1   = (const float*)d_in[2];
  const float* b1   = (const float*)d_in[3];
  const float* W2   = (const float*)d_in[4];
  const float* b2   = (const float*)d_in[5];
  const float* bng  = (const float*)d_in[6];
  const float* bnb  = (const float*)d_in[7];
  const float* bnm  = (const float*)d_in[8];
  const float* bnv  = (const float*)d_in[9];
  const float* W3   = (const float*)d_in[10];
  const float* b3   = (const float*)d_in[11];
  float* out = (float*)d_out;

  float* ws   = (float*)d_ws;
  float* xgap = ws;            // 2048
  float* cw   = ws + 2048;     // 256
  float* A2   = ws + 2304;     // 4096
  float* B2c  = ws + 6400;     // 4096

  const size_t loss_off = (size_t)G_ * B_ * C_ * NSP;  // 25690112

  gap_kernel<<<B_ * C_, 256, 0, stream>>>(x, xgap);
  gate_kernel<<<1, 256, 0, stream>>>(xgap, gat, cw, out + loss_off);
  bn_affine_kernel<<<(E_ * HID_) / 256, 256, 0, stream>>>(b2, bng, bnb, bnm, bnv,
                                                          A2, B2c);
  moe_main<<<B_ * NT, 256, 0, stream>>>(x, W1, b1, W2, W3, b3, cw, A2, B2c, out);
}